// TrustGuardLike_26603027432196
// MI455X (gfx1250) — compile-verified
//
#include <hip/hip_runtime.h>
#include <hip/hip_bf16.h>
#include <math.h>

#define N_NODES 100000
#define NNZ_CNT 3200000
#define E_EDGES 1000000
#define D_IN    256
#define H_DIM   64
#define EDGE_IN 272   // 4*H + 2*8

typedef __attribute__((ext_vector_type(2))) float v2f;
typedef __attribute__((ext_vector_type(8))) float v8f;

// ---------------------------------------------------------------------------
// Small helpers
// ---------------------------------------------------------------------------
__global__ void k_transpose(const float* __restrict__ src, float* __restrict__ dst,
                            int K, int Ncol) {
    int i = blockIdx.x * blockDim.x + threadIdx.x;
    if (i < K * Ncol) {
        int k = i / Ncol, n = i % Ncol;
        dst[n * K + k] = src[i];
    }
}

__global__ void k_zero4(float4* __restrict__ p, int n4) {
    int stride = gridDim.x * blockDim.x;
    for (int i = blockIdx.x * blockDim.x + threadIdx.x; i < n4; i += stride)
        p[i] = make_float4(0.f, 0.f, 0.f, 0.f);
}

// ---------------------------------------------------------------------------
// COO SpMM: out[row] += val * h[col]   (wave per nnz, lane covers 2 of 64 cols)
// row/col/val are wave-uniform -> force scalar (SMEM) loads via readfirstlane.
// ---------------------------------------------------------------------------
__global__ void k_spmm(const int* __restrict__ row, const int* __restrict__ col,
                       const float* __restrict__ val, const float* __restrict__ hsrc,
                       float* __restrict__ out, int nnz) {
    int gw    = (blockIdx.x * blockDim.x + threadIdx.x) >> 5;
    int lane  = threadIdx.x & 31;
    int nwave = (gridDim.x * blockDim.x) >> 5;
    for (int e = gw; e < nnz; e += nwave) {
        int eu = __builtin_amdgcn_readfirstlane(e);
        if (eu + nwave < nnz) {                       // stream-ahead prefetch
            __builtin_prefetch(&row[eu + nwave], 0, 0);
            __builtin_prefetch(&col[eu + nwave], 0, 0);
            __builtin_prefetch(&val[eu + nwave], 0, 0);
        }
        int   r = row[eu];
        int   c = col[eu];
        float w = val[eu];
        const float* hc = hsrc + (size_t)c * H_DIM;
        float h0 = hc[lane];
        float h1 = hc[lane + 32];
        float* orow = out + (size_t)r * H_DIM;
        atomicAdd(&orow[lane],      w * h0);
        atomicAdd(&orow[lane + 32], w * h1);
    }
}

// ---------------------------------------------------------------------------
// Generic  out = tanh(A[ntiles*16 x K] @ W[K x 64] + b)  via WMMA f32 16x16x4
// WT is W transposed: [64][K]. Double-buffered fragment loads to overlap
// VMEM latency with the 4 WMMAs of the previous k-step.
// ---------------------------------------------------------------------------
__global__ void k_gemm_tanh(const float* __restrict__ A, const float* __restrict__ WT,
                            const float* __restrict__ bias, float* __restrict__ out,
                            int K, int ntiles) {
    int gw = (blockIdx.x * blockDim.x + threadIdx.x) >> 5;
    if (gw >= ntiles) return;
    int lane = threadIdx.x & 31;
    int half = lane >> 4, lm = lane & 15;
    int row0 = gw << 4;
    int kh   = half << 1;

    const float* arow = A + (size_t)(row0 + lm) * K;
    const float* br0  = WT + (size_t)(0 * 16 + lm) * K;
    const float* br1  = WT + (size_t)(1 * 16 + lm) * K;
    const float* br2  = WT + (size_t)(2 * 16 + lm) * K;
    const float* br3  = WT + (size_t)(3 * 16 + lm) * K;

    v2f a_c  = *(const v2f*)(arow + kh);
    v2f b_c0 = *(const v2f*)(br0 + kh);
    v2f b_c1 = *(const v2f*)(br1 + kh);
    v2f b_c2 = *(const v2f*)(br2 + kh);
    v2f b_c3 = *(const v2f*)(br3 + kh);

    v8f acc0 = {}, acc1 = {}, acc2 = {}, acc3 = {};
    for (int k0 = 0; k0 < K; k0 += 4) {
        int kn = (k0 + 4 < K) ? (k0 + 4 + kh) : kh;   // wrap keeps loads in-bounds
        v2f a_n  = *(const v2f*)(arow + kn);
        v2f b_n0 = *(const v2f*)(br0 + kn);
        v2f b_n1 = *(const v2f*)(br1 + kn);
        v2f b_n2 = *(const v2f*)(br2 + kn);
        v2f b_n3 = *(const v2f*)(br3 + kn);
        acc0 = __builtin_amdgcn_wmma_f32_16x16x4_f32(false, a_c, false, b_c0, (short)0, acc0, false, false);
        acc1 = __builtin_amdgcn_wmma_f32_16x16x4_f32(false, a_c, false, b_c1, (short)0, acc1, false, false);
        acc2 = __builtin_amdgcn_wmma_f32_16x16x4_f32(false, a_c, false, b_c2, (short)0, acc2, false, false);
        acc3 = __builtin_amdgcn_wmma_f32_16x16x4_f32(false, a_c, false, b_c3, (short)0, acc3, false, false);
        a_c = a_n; b_c0 = b_n0; b_c1 = b_n1; b_c2 = b_n2; b_c3 = b_n3;
    }

    v8f accs[4] = {acc0, acc1, acc2, acc3};
#pragma unroll
    for (int c = 0; c < 4; ++c) {
        int colg = c * 16 + lm;
        float bv = bias[colg];
#pragma unroll
        for (int r = 0; r < 8; ++r) {
            int mrow = row0 + r + (half << 3);
            out[(size_t)mrow * H_DIM + colg] = tanhf(accs[c][r] + bv);
        }
    }
}

// ---------------------------------------------------------------------------
// Gate: t = g*hp + (1-g)*hn,  g = sigmoid([hp|hn] @ Wg + bg)   (K = 128)
// ---------------------------------------------------------------------------
__global__ void k_gate(const float* __restrict__ hp, const float* __restrict__ hn,
                       const float* __restrict__ WTg, const float* __restrict__ bg,
                       float* __restrict__ t, int ntiles) {
    int gw = (blockIdx.x * blockDim.x + threadIdx.x) >> 5;
    if (gw >= ntiles) return;
    int lane = threadIdx.x & 31;
    int half = lane >> 4, lm = lane & 15;
    int row0 = gw << 4;
    int kh   = half << 1;

    const float* br0 = WTg + (size_t)(0 * 16 + lm) * 128;
    const float* br1 = WTg + (size_t)(1 * 16 + lm) * 128;
    const float* br2 = WTg + (size_t)(2 * 16 + lm) * 128;
    const float* br3 = WTg + (size_t)(3 * 16 + lm) * 128;

    auto loadA = [&](int kk) -> v2f {
        const float* s = (kk < 64) ? hp : hn;
        return *(const v2f*)(s + (size_t)(row0 + lm) * H_DIM + (kk & 63));
    };

    v2f a_c  = loadA(kh);
    v2f b_c0 = *(const v2f*)(br0 + kh);
    v2f b_c1 = *(const v2f*)(br1 + kh);
    v2f b_c2 = *(const v2f*)(br2 + kh);
    v2f b_c3 = *(const v2f*)(br3 + kh);

    v8f acc0 = {}, acc1 = {}, acc2 = {}, acc3 = {};
    for (int k0 = 0; k0 < 128; k0 += 4) {
        int kn = (k0 + 4 < 128) ? (k0 + 4 + kh) : kh;
        v2f a_n  = loadA(kn);
        v2f b_n0 = *(const v2f*)(br0 + kn);
        v2f b_n1 = *(const v2f*)(br1 + kn);
        v2f b_n2 = *(const v2f*)(br2 + kn);
        v2f b_n3 = *(const v2f*)(br3 + kn);
        acc0 = __builtin_amdgcn_wmma_f32_16x16x4_f32(false, a_c, false, b_c0, (short)0, acc0, false, false);
        acc1 = __builtin_amdgcn_wmma_f32_16x16x4_f32(false, a_c, false, b_c1, (short)0, acc1, false, false);
        acc2 = __builtin_amdgcn_wmma_f32_16x16x4_f32(false, a_c, false, b_c2, (short)0, acc2, false, false);
        acc3 = __builtin_amdgcn_wmma_f32_16x16x4_f32(false, a_c, false, b_c3, (short)0, acc3, false, false);
        a_c = a_n; b_c0 = b_n0; b_c1 = b_n1; b_c2 = b_n2; b_c3 = b_n3;
    }

    v8f accs[4] = {acc0, acc1, acc2, acc3};
#pragma unroll
    for (int c = 0; c < 4; ++c) {
        int colg = c * 16 + lm;
        float bv = bg[colg];
#pragma unroll
        for (int r = 0; r < 8; ++r) {
            int mrow = row0 + r + (half << 3);
            size_t idx = (size_t)mrow * H_DIM + colg;
            float g = 1.f / (1.f + expf(-(accs[c][r] + bv)));
            t[idx] = g * hp[idx] + (1.f - g) * hn[idx];
        }
    }
}

// ---------------------------------------------------------------------------
// Edge scorer: fused feature build + [E x 272]@[272 x 64] (WMMA, LDS-staged)
// + relu + 64->1 dot.  2 waves / block, 16 edges / wave.
// ---------------------------------------------------------------------------
__global__ __launch_bounds__(64) void k_edge(
    const float* __restrict__ h, const int* __restrict__ u, const int* __restrict__ v,
    const float* __restrict__ ets, const float* __restrict__ WTe1,
    const float* __restrict__ be1, const float* __restrict__ we2,
    const float* __restrict__ be2, float* __restrict__ out) {

    __shared__ __align__(16) float feat[2][16][EDGE_IN];
    __shared__ __align__(16) float hid[2][16][H_DIM];
    __shared__ int suv[2][2][16];

    int w    = threadIdx.x >> 5;
    int lane = threadIdx.x & 31;
    int tile = blockIdx.x * 2 + w;
    int e0   = tile << 4;
    int half = lane >> 4, lm = lane & 15;
    int kh   = half << 1;

    if (lane < 16) {
        suv[w][0][lane] = u[e0 + lane];
        suv[w][1][lane] = v[e0 + lane];
    }
    __syncthreads();

    // hu | hv | |hu-hv| | hu*hv   (float4 gathers -> ds_store_b128)
    for (int i = lane; i < 16 * 16; i += 32) {
        int r = i >> 4, c = (i & 15) << 2;
        float4 a4 = *(const float4*)(h + (size_t)suv[w][0][r] * H_DIM + c);
        float4 b4 = *(const float4*)(h + (size_t)suv[w][1][r] * H_DIM + c);
        *(float4*)&feat[w][r][c]      = a4;
        *(float4*)&feat[w][r][64 + c] = b4;
        float4 d4 = make_float4(fabsf(a4.x - b4.x), fabsf(a4.y - b4.y),
                                fabsf(a4.z - b4.z), fabsf(a4.w - b4.w));
        float4 p4 = make_float4(a4.x * b4.x, a4.y * b4.y, a4.z * b4.z, a4.w * b4.w);
        *(float4*)&feat[w][r][128 + c] = d4;
        *(float4*)&feat[w][r][192 + c] = p4;
    }
    // time features
    if (lane < 16) {
        float t = ets[e0 + lane];
        t = fminf(fmaxf(t, 0.f), 1.f);
#pragma unroll
        for (int f = 0; f < 8; ++f) {
            float ang = t * 3.14159265358979323846f * (float)(f + 1);
            feat[w][lane][256 + f] = sinf(ang);
            feat[w][lane][264 + f] = cosf(ang);
        }
    }
    __syncthreads();

    const float* frow = &feat[w][lm][0];
    const float* br0  = WTe1 + (size_t)(0 * 16 + lm) * EDGE_IN;
    const float* br1  = WTe1 + (size_t)(1 * 16 + lm) * EDGE_IN;
    const float* br2  = WTe1 + (size_t)(2 * 16 + lm) * EDGE_IN;
    const float* br3  = WTe1 + (size_t)(3 * 16 + lm) * EDGE_IN;

    v2f a_c  = *(const v2f*)(frow + kh);              // ds_load_b64
    v2f b_c0 = *(const v2f*)(br0 + kh);
    v2f b_c1 = *(const v2f*)(br1 + kh);
    v2f b_c2 = *(const v2f*)(br2 + kh);
    v2f b_c3 = *(const v2f*)(br3 + kh);

    v8f acc0 = {}, acc1 = {}, acc2 = {}, acc3 = {};
    for (int k0 = 0; k0 < EDGE_IN; k0 += 4) {
        int kn = (k0 + 4 < EDGE_IN) ? (k0 + 4 + kh) : kh;
        v2f a_n  = *(const v2f*)(frow + kn);
        v2f b_n0 = *(const v2f*)(br0 + kn);
        v2f b_n1 = *(const v2f*)(br1 + kn);
        v2f b_n2 = *(const v2f*)(br2 + kn);
        v2f b_n3 = *(const v2f*)(br3 + kn);
        acc0 = __builtin_amdgcn_wmma_f32_16x16x4_f32(false, a_c, false, b_c0, (short)0, acc0, false, false);
        acc1 = __builtin_amdgcn_wmma_f32_16x16x4_f32(false, a_c, false, b_c1, (short)0, acc1, false, false);
        acc2 = __builtin_amdgcn_wmma_f32_16x16x4_f32(false, a_c, false, b_c2, (short)0, acc2, false, false);
        acc3 = __builtin_amdgcn_wmma_f32_16x16x4_f32(false, a_c, false, b_c3, (short)0, acc3, false, false);
        a_c = a_n; b_c0 = b_n0; b_c1 = b_n1; b_c2 = b_n2; b_c3 = b_n3;
    }

    v8f accs[4] = {acc0, acc1, acc2, acc3};
#pragma unroll
    for (int c = 0; c < 4; ++c) {
        int colg = c * 16 + lm;
        float bv = be1[colg];
#pragma unroll
        for (int r = 0; r < 8; ++r)
            hid[w][r + (half << 3)][colg] = fmaxf(accs[c][r] + bv, 0.f);
    }
    __syncthreads();

    if (lane < 16) {
        float s = be2[0];
        for (int j = 0; j < H_DIM; ++j)
            s += hid[w][lane][j] * we2[j];
        out[e0 + lane] = s;
    }
}

// ---------------------------------------------------------------------------
extern "C" void kernel_launch(void* const* d_in, const int* in_sizes, int n_in,
                              void* d_out, int out_size, void* d_ws, size_t ws_size,
                              hipStream_t stream) {
    const float* x        = (const float*)d_in[0];
    const int*   pos_row  = (const int*)d_in[1];
    const int*   pos_col  = (const int*)d_in[2];
    const float* pos_val  = (const float*)d_in[3];
    const int*   neg_row  = (const int*)d_in[4];
    const int*   neg_col  = (const int*)d_in[5];
    const float* neg_val  = (const float*)d_in[6];
    const int*   u        = (const int*)d_in[7];
    const int*   v        = (const int*)d_in[8];
    const float* ets      = (const float*)d_in[9];
    const float* W_in     = (const float*)d_in[10];
    const float* b_in     = (const float*)d_in[11];
    const float* W_gate   = (const float*)d_in[12];
    const float* b_gate   = (const float*)d_in[13];
    const float* W_out    = (const float*)d_in[14];
    const float* b_out    = (const float*)d_in[15];
    const float* W_e1     = (const float*)d_in[16];
    const float* b_e1     = (const float*)d_in[17];
    const float* W_e2     = (const float*)d_in[18];
    const float* b_e2     = (const float*)d_in[19];
    float* out = (float*)d_out;

    const size_t NH = (size_t)N_NODES * H_DIM;
    float* ws   = (float*)d_ws;
    float* h    = ws;
    float* hp   = h    + NH;
    float* hn   = hp   + NH;
    float* t    = hn   + NH;
    float* WTin = t    + NH;                 // 64 x 256
    float* WTg  = WTin + 64 * 256;           // 2 x (64 x 128)
    float* WTo  = WTg  + 2 * 64 * 128;       // 2 x (64 x 64)
    float* WTe1 = WTo  + 2 * 64 * 64;        // 64 x 272

    // weight transposes (K-major for B-fragment float2 loads)
    k_transpose<<<(256 * 64 + 255) / 256, 256, 0, stream>>>(W_in, WTin, 256, 64);
    for (int hop = 0; hop < 2; ++hop) {
        k_transpose<<<(128 * 64 + 255) / 256, 256, 0, stream>>>(
            W_gate + (size_t)hop * 128 * 64, WTg + (size_t)hop * 64 * 128, 128, 64);
        k_transpose<<<(64 * 64 + 255) / 256, 256, 0, stream>>>(
            W_out + (size_t)hop * 64 * 64, WTo + (size_t)hop * 64 * 64, 64, 64);
    }
    k_transpose<<<(EDGE_IN * 64 + 255) / 256, 256, 0, stream>>>(W_e1, WTe1, EDGE_IN, 64);

    const int ntiles = N_NODES / 16;                  // 6250
    const int ggrid  = (ntiles + 7) / 8;              // 8 waves / 256-thread block

    // h = tanh(x @ W_in + b_in)
    k_gemm_tanh<<<ggrid, 256, 0, stream>>>(x, WTin, b_in, h, D_IN, ntiles);

    for (int hop = 0; hop < 2; ++hop) {
        // hp, hn are contiguous: zero both at once
        k_zero4<<<2048, 256, 0, stream>>>((float4*)hp, (int)(2 * NH / 4));
        k_spmm<<<4096, 256, 0, stream>>>(pos_row, pos_col, pos_val, h, hp, NNZ_CNT);
        k_spmm<<<4096, 256, 0, stream>>>(neg_row, neg_col, neg_val, h, hn, NNZ_CNT);
        k_gate<<<ggrid, 256, 0, stream>>>(hp, hn, WTg + (size_t)hop * 64 * 128,
                                          b_gate + (size_t)hop * 64, t, ntiles);
        k_gemm_tanh<<<ggrid, 256, 0, stream>>>(t, WTo + (size_t)hop * 64 * 64,
                                               b_out + (size_t)hop * 64, h, H_DIM, ntiles);
    }

    // edge scorer: 62500 tiles, 2 waves / block
    const int etiles = E_EDGES / 16;                  // 62500
    k_edge<<<etiles / 2, 64, 0, stream>>>(h, u, v, ets, WTe1, b_e1, W_e2, b_e2, out);
}